// BWCaster_86337432584570
// MI455X (gfx1250) — compile-verified
//
#include <hip/hip_runtime.h>

// ---------------------------------------------------------------------------
// TensoRF-style skinning weights on MI455X (gfx1250).
//   Phase 1: per-joint 4x4 transform of points via V_WMMA_F32_16X16X4_F32
//            (A = 16 transform rows x K=4, B = 4 x 16 points), staged to LDS.
//   Phase 2: per-thread gather/interp (bilinear plane * linear line, dot over
//            C=16), ReLU, store [N,J].
// Planes (75 MB) are L2-resident on the 192 MB L2; kernel is gather-latency
// bound, so we keep LDS at ~37 KB/block (8 blocks/WGP) and registers modest.
// ---------------------------------------------------------------------------

typedef float v2f __attribute__((ext_vector_type(2)));
typedef float v8f __attribute__((ext_vector_type(8)));

#ifndef __has_builtin
#define __has_builtin(x) 0
#endif
#if __has_builtin(__builtin_amdgcn_wmma_f32_16x16x4_f32)
#define USE_WMMA_XFORM 1
#else
#define USE_WMMA_XFORM 0
#endif

constexpr int JN = 24;          // joints
constexpr int CN = 16;          // channels
constexpr int GN = 128;         // grid resolution (planes GxG, lines G)
constexpr int PTS_PER_BLOCK = 128;
constexpr int LDS_STRIDE = 73;  // 72 coords + 1 pad -> odd stride, no bank conflicts

__global__ void __launch_bounds__(PTS_PER_BLOCK)
tensorf_weights_kernel(const float* __restrict__ xyz,
                       const float* __restrict__ transforms,
                       const float* __restrict__ plane0,
                       const float* __restrict__ plane1,
                       const float* __restrict__ plane2,
                       const float* __restrict__ line0,
                       const float* __restrict__ line1,
                       const float* __restrict__ line2,
                       float* __restrict__ out, int N)
{
    __shared__ float lds_pts[PTS_PER_BLOCK][LDS_STRIDE];

    const int tid       = threadIdx.x;
    const int lane      = tid & 31;
    const int wave      = tid >> 5;
    const int blockBase = blockIdx.x * PTS_PER_BLOCK;

    // ---------------- Phase 1: transforms -> LDS ---------------------------
#if USE_WMMA_XFORM
    {
        // A tiles: 6 WMMAs cover 96 rows = 24 joints x 4 matrix rows.
        // A 16x4 f32 layout: lanes 0-15 and 16-31 both hold rows M=0..15;
        // lanes<16 carry K=0,1 and lanes>=16 carry K=2,3.
        const int rowM = lane & 15;
        const int k0   = (lane < 16) ? 0 : 2;
        v2f aT[6];
#pragma unroll
        for (int t = 0; t < 6; ++t) {
            const int gr   = t * 16 + rowM;          // global row 0..95
            const int jA   = gr >> 2;                // joint
            const int trA  = gr & 3;                 // row of 4x4
            const float* tb = transforms + jA * 16 + trA * 4 + k0;
            aT[t].x = tb[0];
            aT[t].y = tb[1];
        }

        // Two batches of 16 points per wave (wave owns 32 points).
#pragma unroll
        for (int r = 0; r < 2; ++r) {
            const int pl = lane & 15;
            const int plocal = wave * 32 + r * 16 + pl;
            const int pg = min(blockBase + plocal, N - 1);
            const float bx = xyz[pg * 3 + 0];
            const float by = xyz[pg * 3 + 1];
            const float bz = xyz[pg * 3 + 2];
            // B 4x16 f32: lanes 0-15 hold (K=0,K=1)=(x,y); lanes 16-31 hold
            // (K=2,K=3)=(z,1) for the same 16 columns (points).
            v2f b;
            b.x = (lane < 16) ? bx : bz;
            b.y = (lane < 16) ? by : 1.0f;

#pragma unroll
            for (int t = 0; t < 6; ++t) {
                v8f c = {0.f, 0.f, 0.f, 0.f, 0.f, 0.f, 0.f, 0.f};
                v8f d = __builtin_amdgcn_wmma_f32_16x16x4_f32(
                    false, aT[t], false, b, (short)0, c, false, false);
                // D 16x16 f32: lane = column (point), vgpr i -> row i (+8 for
                // upper half-wave). Row -> (joint, matrix-row); keep rows 0..2.
                const int jadd = (lane < 16) ? 0 : 2;
#pragma unroll
                for (int r8 = 0; r8 < 8; ++r8) {
                    const int trow = r8 & 3;
                    if (trow < 3) {
                        const int j = t * 4 + jadd + (r8 >> 2);
                        lds_pts[plocal][j * 3 + trow] = d[r8];
                    }
                }
            }
        }
    }
#else
    {
        // Scalar fallback: each thread transforms its own point.
        const int pg = min(blockBase + tid, N - 1);
        const float x = xyz[pg * 3 + 0];
        const float y = xyz[pg * 3 + 1];
        const float z = xyz[pg * 3 + 2];
#pragma unroll 4
        for (int j = 0; j < JN; ++j) {
            const float* T = transforms + j * 16;
            lds_pts[tid][j * 3 + 0] = T[0] * x + T[1] * y + T[2]  * z + T[3];
            lds_pts[tid][j * 3 + 1] = T[4] * x + T[5] * y + T[6]  * z + T[7];
            lds_pts[tid][j * 3 + 2] = T[8] * x + T[9] * y + T[10] * z + T[11];
        }
    }
#endif
    __syncthreads();

    // ---------------- Phase 2: gather + interpolate ------------------------
    const int   p      = blockBase + tid;
    const bool  active = p < N;
    const float* planesArr[3] = {plane0, plane1, plane2};
    const float* linesArr[3]  = {line0, line1, line2};
    const int m0s[3] = {0, 0, 1};   // MAT_MODE x
    const int m1s[3] = {1, 2, 2};   // MAT_MODE y
    const int vs[3]  = {2, 1, 0};   // VEC_MODE

    for (int j = 0; j < JN; ++j) {
        float pt[3];
        pt[0] = lds_pts[tid][j * 3 + 0];
        pt[1] = lds_pts[tid][j * 3 + 1];
        pt[2] = lds_pts[tid][j * 3 + 2];

        // Per-axis sample prep. g = (pts - AABB_LO)*INV_SIZE - 1 = (2/3)*pts.
        float w0v[3], w1v[3];
        int   ic0[3], ic1[3];
#pragma unroll
        for (int a = 0; a < 3; ++a) {
            const float fi = (pt[a] * (2.0f / 3.0f) + 1.0f) * 0.5f * (GN - 1);
            const float fl = floorf(fi);
            const int   i0 = (int)fl;
            const float w1 = fi - fl;
            w0v[a] = (i0 >= 0 && i0 < GN)      ? (1.0f - w1) : 0.0f;
            w1v[a] = (i0 >= -1 && i0 < GN - 1) ? w1          : 0.0f;
            ic0[a] = min(max(i0, 0), GN - 1);
            ic1[a] = min(max(i0 + 1, 0), GN - 1);
        }

        float sigma = 0.0f;
#pragma unroll
        for (int m = 0; m < 3; ++m) {
            const int m0 = m0s[m], m1 = m1s[m], v = vs[m];
            const float wx0 = w0v[m0], wx1 = w1v[m0];
            const float wy0 = w0v[m1], wy1 = w1v[m1];
            const float w00 = wx0 * wy0, w01 = wx1 * wy0;
            const float w10 = wx0 * wy1, w11 = wx1 * wy1;
            const int r0 = ic0[m1] * GN, r1 = ic1[m1] * GN;
            const int x0 = ic0[m0], x1 = ic1[m0];
            const int z0 = ic0[v],  z1 = ic1[v];
            const float wz0 = w0v[v], wz1 = w1v[v];
            const float* __restrict__ pb = planesArr[m] + (size_t)j * CN * GN * GN;
            const float* __restrict__ lb = linesArr[m]  + (size_t)j * CN * GN;
#pragma unroll 4
            for (int c = 0; c < CN; ++c) {
                const float* __restrict__ pc = pb + c * (GN * GN);
                const float* __restrict__ lc = lb + c * GN;
                const float bil = pc[r0 + x0] * w00 + pc[r0 + x1] * w01 +
                                  pc[r1 + x0] * w10 + pc[r1 + x1] * w11;
                const float lf  = lc[z0] * wz0 + lc[z1] * wz1;
                sigma += bil * lf;
            }
        }
        if (active) out[(size_t)p * JN + j] = fmaxf(sigma, 0.0f);
    }
}

extern "C" void kernel_launch(void* const* d_in, const int* in_sizes, int n_in,
                              void* d_out, int out_size, void* d_ws, size_t ws_size,
                              hipStream_t stream)
{
    (void)n_in; (void)out_size; (void)d_ws; (void)ws_size;
    const float* xyz        = (const float*)d_in[0];
    const float* transforms = (const float*)d_in[1];
    const float* plane0     = (const float*)d_in[2];
    const float* plane1     = (const float*)d_in[3];
    const float* plane2     = (const float*)d_in[4];
    const float* line0      = (const float*)d_in[5];
    const float* line1      = (const float*)d_in[6];
    const float* line2      = (const float*)d_in[7];
    float* out = (float*)d_out;

    const int N = in_sizes[0] / 3;   // xyz is [N,3]
    const int nblocks = (N + PTS_PER_BLOCK - 1) / PTS_PER_BLOCK;
    tensorf_weights_kernel<<<nblocks, PTS_PER_BLOCK, 0, stream>>>(
        xyz, transforms, plane0, plane1, plane2, line0, line1, line2, out, N);
}